// GraphTransformerGFN_83794811945133
// MI455X (gfx1250) — compile-verified
//
#include <hip/hip_runtime.h>

// ---------------------------------------------------------------------------
// GraphTransformerGFN forward for MI455X (gfx1250, wave32).
// Dense linears run on V_WMMA_F32_16X16X4_F32 (fp32 matrix pipe — keeps the
// fp32 reference accuracy; weights are L2-resident, problem is matrix-compute
// bound at ~0.6 TFLOP). Graph scatter ops use fp32 atomics.
//
// GEMM: 128x64 block tile, 8 waves, each wave owns 32x32 (4 accumulators).
// B is stored transposed in LDS (row stride 34 floats -> 8B-aligned b64
// fragment loads, no repacking movs). Next k-tile prefetched with
// global_prefetch_b8.
// ---------------------------------------------------------------------------

typedef __attribute__((ext_vector_type(2))) float v2f;
typedef __attribute__((ext_vector_type(8))) float v8f;

static constexpr int kNodes  = 49152;
static constexpr int kGraphs = 1536;
static constexpr int kEdges  = 196608;
static constexpr int kNonEd  = 98304;
static constexpr int kXDim   = 74;
static constexpr int kEDim   = 4;
static constexpr int kGDim   = 32;
static constexpr int kEmb    = 128;
static constexpr int kNAug   = kNodes + kGraphs;    // 50688
static constexpr int kEPre   = kEdges + 2 * kNodes; // 294912
static constexpr int kEAug   = kEPre + kNAug;       // 345600
static constexpr float kSlope = 0.01f;

// ---------------------------------------------------------------------------
// WMMA fp32 GEMM:  C[M,N] = act(A[M,K] @ W[K,N] + bias)  (+= C if ACC)
// Requirements (hold at every call site): M % 128 == 0, N % 64 == 0.
// K zero-padded in LDS (handles K = 4, 32, 74, 128, 256, 512).
// ---------------------------------------------------------------------------
template <int ACT, bool ACC>
__global__ __launch_bounds__(256) void gemm_wmma_f32(
    const float* __restrict__ A, const float* __restrict__ W,
    const float* __restrict__ bias, float* __restrict__ C,
    int M, int K, int N) {
  __shared__ float As[128][34];  // [m][k], stride 34 -> 8B aligned rows
  __shared__ float Bs[64][34];   // [n][k] (transposed), same stride

  const int tid  = threadIdx.x;
  const int lane = tid & 31;
  const int wv   = tid >> 5;
  const int wm   = wv >> 1;            // 0..3 -> 32-row slab
  const int wn   = wv & 1;             // 0..1 -> 32-col slab
  const int m0   = blockIdx.y << 7;
  const int n0   = blockIdx.x << 6;
  const int kh   = (lane >> 4) << 1;   // lanes 0-15 -> k+0,1 ; 16-31 -> k+2,3
  const int l16  = lane & 15;

  v8f acc00 = {}, acc01 = {}, acc10 = {}, acc11 = {};

  // global->LDS mapping
  const int ar = tid >> 1;             // A row 0..127
  const int ac = (tid & 1) << 4;       // A col base (16 floats per thread)
  const int br = tid >> 3;             // W row (k) 0..31
  const int bc = (tid & 7) << 3;       // W col base (8 floats per thread)

  const float* arow0 = &As[(wm << 5) + l16][0];
  const float* arow1 = &As[(wm << 5) + 16 + l16][0];
  const float* brow0 = &Bs[(wn << 5) + l16][0];
  const float* brow1 = &Bs[(wn << 5) + 16 + l16][0];

  for (int k0 = 0; k0 < K; k0 += 32) {
    const float* ap = A + (size_t)(m0 + ar) * K + (k0 + ac);
    const float* wp = W + (size_t)(k0 + br) * N + (n0 + bc);
    if (k0 + 32 < K) {  // prefetch next k-tile (global_prefetch_b8)
      __builtin_prefetch(ap + 32, 0, 0);
      __builtin_prefetch(wp + (size_t)32 * N, 0, 0);
    }
#pragma unroll
    for (int i = 0; i < 16; ++i)
      As[ar][ac + i] = (k0 + ac + i < K) ? ap[i] : 0.0f;
    {
      const bool kok = (k0 + br) < K;
#pragma unroll
      for (int i = 0; i < 8; ++i)
        Bs[bc + i][br] = kok ? wp[i] : 0.0f;  // transpose on store
    }
    __syncthreads();
#pragma unroll
    for (int kk = 0; kk < 32; kk += 4) {
      const int ka = kk + kh;
      const v2f a0 = *(const v2f*)(arow0 + ka);
      const v2f a1 = *(const v2f*)(arow1 + ka);
      const v2f b0 = *(const v2f*)(brow0 + ka);
      const v2f b1 = *(const v2f*)(brow1 + ka);
      acc00 = __builtin_amdgcn_wmma_f32_16x16x4_f32(false, a0, false, b0,
                                                    (short)0, acc00, false, false);
      acc01 = __builtin_amdgcn_wmma_f32_16x16x4_f32(false, a0, false, b1,
                                                    (short)0, acc01, false, false);
      acc10 = __builtin_amdgcn_wmma_f32_16x16x4_f32(false, a1, false, b0,
                                                    (short)0, acc10, false, false);
      acc11 = __builtin_amdgcn_wmma_f32_16x16x4_f32(false, a1, false, b1,
                                                    (short)0, acc11, false, false);
    }
    __syncthreads();
  }

  const int rbase = (lane >> 4) << 3;  // C layout: lanes 16-31 hold M=8..15
#pragma unroll
  for (int ms = 0; ms < 2; ++ms) {
    const int mrow = m0 + (wm << 5) + (ms << 4) + rbase;
#pragma unroll
    for (int ns = 0; ns < 2; ++ns) {
      const int nc = n0 + (wn << 5) + (ns << 4) + l16;
      const float bv = bias[nc];
      const v8f acc = ms == 0 ? (ns == 0 ? acc00 : acc01)
                              : (ns == 0 ? acc10 : acc11);
#pragma unroll
      for (int r = 0; r < 8; ++r) {
        const size_t idx = (size_t)(mrow + r) * N + nc;
        float v = acc[r] + bv;
        if (ACT == 1) v = (v > 0.0f) ? v : kSlope * v;
        if (ACC) v += C[idx];
        C[idx] = v;
      }
    }
  }
}

// ---------------------------------------------------------------------------
// Small-N heads: one wave per row, N <= 8.
// ---------------------------------------------------------------------------
template <int N>
__global__ void rowdot_kernel(const float* __restrict__ A,
                              const float* __restrict__ W,
                              const float* __restrict__ bias,
                              float* __restrict__ C, int M, int K) {
  const int row  = (int)((blockIdx.x * (long long)blockDim.x + threadIdx.x) >> 5);
  const int lane = threadIdx.x & 31;
  if (row >= M) return;
  float acc[N];
#pragma unroll
  for (int j = 0; j < N; ++j) acc[j] = 0.0f;
  for (int k = lane; k < K; k += 32) {
    const float a = A[(size_t)row * K + k];
#pragma unroll
    for (int j = 0; j < N; ++j) acc[j] += a * W[(size_t)k * N + j];
  }
#pragma unroll
  for (int j = 0; j < N; ++j) {
    float v = acc[j];
    for (int off = 16; off; off >>= 1) v += __shfl_xor(v, off, 32);
    if (lane == 0) C[(size_t)row * N + j] = v + bias[j];
  }
}

// (o[ia[p*sa]] + o[ib[p*sb]]) @ W[128,N] + b  — wave per pair.
template <int N>
__global__ void pairdot_kernel(const float* __restrict__ o,
                               const int* __restrict__ ia, int sa,
                               const int* __restrict__ ib, int sb,
                               const float* __restrict__ W,
                               const float* __restrict__ bias,
                               float* __restrict__ C, int M) {
  const int p    = (int)((blockIdx.x * (long long)blockDim.x + threadIdx.x) >> 5);
  const int lane = threadIdx.x & 31;
  if (p >= M) return;
  const float* ra = o + (size_t)ia[(size_t)p * sa] * kEmb;
  const float* rb = o + (size_t)ib[(size_t)p * sb] * kEmb;
  float acc[N];
#pragma unroll
  for (int j = 0; j < N; ++j) acc[j] = 0.0f;
#pragma unroll
  for (int q4 = 0; q4 < 4; ++q4) {
    const int k = lane + q4 * 32;
    const float a = ra[k] + rb[k];
#pragma unroll
    for (int j = 0; j < N; ++j) acc[j] += a * W[k * N + j];
  }
#pragma unroll
  for (int j = 0; j < N; ++j) {
    float v = acc[j];
    for (int off = 16; off; off >>= 1) v += __shfl_xor(v, off, 32);
    if (lane == 0) C[(size_t)p * N + j] = v + bias[j];
  }
}

// ---------------------------------------------------------------------------
// Elementwise / graph kernels
// ---------------------------------------------------------------------------
__global__ void fill_f32_kernel(float* p, float v, long long n) {
  long long i = (long long)blockIdx.x * blockDim.x + threadIdx.x;
  if (i < n) p[i] = v;
}
__global__ void fill_u32_kernel(unsigned* p, unsigned v, long long n) {
  long long i = (long long)blockIdx.x * blockDim.x + threadIdx.x;
  if (i < n) p[i] = v;
}
__global__ void add_inplace_kernel(float* __restrict__ d,
                                   const float* __restrict__ s, long long n) {
  long long i = (long long)blockIdx.x * blockDim.x + threadIdx.x;
  if (i < n) d[i] += s[i];
}

__global__ void build_edges_kernel(const int* __restrict__ eidx,
                                   const int* __restrict__ batch,
                                   int* __restrict__ esrc, int* __restrict__ edst) {
  int i = blockIdx.x * blockDim.x + threadIdx.x;
  if (i < kEdges) { esrc[i] = eidx[i]; edst[i] = eidx[kEdges + i]; }
  if (i < kNodes) {
    int v = batch[i] + kNodes;
    esrc[kEdges + i] = i;          edst[kEdges + i] = v;
    esrc[kEdges + kNodes + i] = v; edst[kEdges + kNodes + i] = i;
  }
  if (i < kNAug) { esrc[kEPre + i] = i; edst[kEPre + i] = i; }
}

__global__ void init_ep_kernel(float* __restrict__ ea) {
  long long i = (long long)blockIdx.x * blockDim.x + threadIdx.x;
  if (i >= (long long)2 * kNodes * kEmb) return;
  ea[(size_t)kEdges * kEmb + i] = ((i & 127) == 0) ? 1.0f : 0.0f;
}

__global__ void loopattr_sum_kernel(const float* __restrict__ ea,
                                    const int* __restrict__ edst,
                                    float* __restrict__ s, float* __restrict__ cnt) {
  long long i = (long long)blockIdx.x * blockDim.x + threadIdx.x;
  if (i >= (long long)kEPre * kEmb) return;
  const int e = (int)(i >> 7), f = (int)(i & 127);
  const int d = edst[e];
  atomicAdd(&s[(size_t)d * kEmb + f], ea[i]);
  if (f == 0) atomicAdd(&cnt[d], 1.0f);
}
__global__ void loopattr_write_kernel(const float* __restrict__ s,
                                      const float* __restrict__ cnt,
                                      float* __restrict__ ea) {
  long long i = (long long)blockIdx.x * blockDim.x + threadIdx.x;
  if (i >= (long long)kNAug * kEmb) return;
  const int r = (int)(i >> 7);
  ea[(size_t)(kEPre + r) * kEmb + (i & 127)] = s[i] / fmaxf(cnt[r], 1.0f);
}

// Per-graph LayerNorm over 33 rows x 128 feats (32 nodes + 1 context row).
__global__ __launch_bounds__(128) void graph_ln_kernel(const float* __restrict__ x,
                                                       float* __restrict__ y) {
  const int g = blockIdx.x, t = threadIdx.x;
  float s = 0.0f, ss = 0.0f;
  for (int r = 0; r < 33; ++r) {
    const int row = (r < 32) ? (g * 32 + r) : (kNodes + g);
    const float v = x[(size_t)row * kEmb + t];
    s += v; ss += v * v;
  }
  __shared__ float red[128];
  red[t] = s; __syncthreads();
  for (int st = 64; st; st >>= 1) { if (t < st) red[t] += red[t + st]; __syncthreads(); }
  const float mean = red[0] / (33.0f * 128.0f);
  __syncthreads();
  red[t] = ss; __syncthreads();
  for (int st = 64; st; st >>= 1) { if (t < st) red[t] += red[t + st]; __syncthreads(); }
  const float var = red[0] / (33.0f * 128.0f) - mean * mean;
  const float inv = rsqrtf(var + 1e-5f);
  for (int r = 0; r < 33; ++r) {
    const int row = (r < 32) ? (g * 32 + r) : (kNodes + g);
    y[(size_t)row * kEmb + t] = (x[(size_t)row * kEmb + t] - mean) * inv;
  }
}

__global__ void msg_agg_kernel(const float* __restrict__ on,
                               const float* __restrict__ ea,
                               const int* __restrict__ esrc,
                               const int* __restrict__ edst,
                               float* __restrict__ agg) {
  long long i = (long long)blockIdx.x * blockDim.x + threadIdx.x;
  if (i >= (long long)kEAug * kEmb) return;
  const int e = (int)(i >> 7), f = (int)(i & 127);
  float m = on[(size_t)esrc[e] * kEmb + f] + ea[i];
  m = fmaxf(m, 0.0f) + 1e-7f;
  atomicAdd(&agg[(size_t)edst[e] * kEmb + f], m);
}

__global__ void concat_kernel(const float* __restrict__ a,
                              const float* __restrict__ b,
                              float* __restrict__ xc) {
  long long i = (long long)blockIdx.x * blockDim.x + threadIdx.x;
  if (i >= (long long)kNAug * 256) return;
  const int r = (int)(i >> 8), c = (int)(i & 255);
  xc[i] = (c < 128) ? a[(size_t)r * kEmb + c] : b[(size_t)r * kEmb + (c - 128)];
}

__device__ __forceinline__ unsigned fenc(float f) {
  unsigned u = __float_as_uint(f);
  return (u & 0x80000000u) ? ~u : (u | 0x80000000u);
}
__device__ __forceinline__ float fdec(unsigned u) {
  return __uint_as_float((u & 0x80000000u) ? (u & 0x7FFFFFFFu) : ~u);
}

__global__ void attn_logit_kernel(const float* __restrict__ q,
                                  const float* __restrict__ k,
                                  const float* __restrict__ ee,
                                  const int* __restrict__ esrc,
                                  const int* __restrict__ edst,
                                  float* __restrict__ logits) {
  const int idx  = (int)((blockIdx.x * (long long)blockDim.x + threadIdx.x) >> 5);
  const int lane = threadIdx.x & 31;
  if (idx >= kEAug * 2) return;
  const int e = idx >> 1, h = idx & 1;
  const float* qp = q + (size_t)edst[e] * 256 + h * 128;
  const float* kp = k + (size_t)esrc[e] * 256 + h * 128;
  const float* ep = ee + (size_t)e * 256 + h * 128;
  float acc = 0.0f;
#pragma unroll
  for (int j = 0; j < 4; ++j) {
    const int kk = lane + j * 32;
    acc += qp[kk] * (kp[kk] + ep[kk]);
  }
  for (int off = 16; off; off >>= 1) acc += __shfl_xor(acc, off, 32);
  if (lane == 0) logits[idx] = acc * 0.08838834764831845f;  // 1/sqrt(128)
}

__global__ void seg_max_kernel(const float* __restrict__ logits,
                               const int* __restrict__ edst,
                               unsigned* __restrict__ mx) {
  long long i = (long long)blockIdx.x * blockDim.x + threadIdx.x;
  if (i >= (long long)kEAug * 2) return;
  const int e = (int)(i >> 1), h = (int)(i & 1);
  atomicMax(&mx[(size_t)edst[e] * 2 + h], fenc(logits[i]));
}

// out[dst] += exp(l - m) * (v[src] + ee)  ;  ssum[dst,h] += exp(l - m)
__global__ void attn_accum_kernel(const float* __restrict__ logits,
                                  const unsigned* __restrict__ mx,
                                  const float* __restrict__ v,
                                  const float* __restrict__ ee,
                                  const int* __restrict__ esrc,
                                  const int* __restrict__ edst,
                                  float* __restrict__ ssum,
                                  float* __restrict__ out) {
  long long i = (long long)blockIdx.x * blockDim.x + threadIdx.x;
  if (i >= (long long)kEAug * 256) return;
  const int e = (int)(i >> 8), c = (int)(i & 255);
  const int h = c >> 7, f = c & 127;
  const int d = edst[e];
  const float w = expf(logits[(size_t)e * 2 + h] - fdec(mx[(size_t)d * 2 + h]));
  if (f == 0) atomicAdd(&ssum[(size_t)d * 2 + h], w);
  const float val = v[(size_t)esrc[e] * 256 + c] + ee[(size_t)e * 256 + c];
  atomicAdd(&out[(size_t)d * 256 + c], w * val);
}

__global__ void attn_norm_kernel(float* __restrict__ out,
                                 const float* __restrict__ ssum) {
  long long i = (long long)blockIdx.x * blockDim.x + threadIdx.x;
  if (i >= (long long)kNAug * 256) return;
  const int r = (int)(i >> 8), h = (int)((i & 255) >> 7);
  out[i] /= (ssum[(size_t)r * 2 + h] + 1e-16f);
}

// o += l_h * cs[:, :128] + cs[:, 128:]   (cs gathered via aug_batch)
__global__ void o_update_kernel(float* __restrict__ o,
                                const float* __restrict__ lh,
                                const float* __restrict__ cs) {
  long long i = (long long)blockIdx.x * blockDim.x + threadIdx.x;
  if (i >= (long long)kNAug * kEmb) return;
  const int r = (int)(i >> 7), f = (int)(i & 127);
  const int gb = (r < kNodes) ? (r >> 5) : (r - kNodes);
  const float* c = cs + (size_t)gb * 256;
  o[i] += lh[i] * c[f] + c[128 + f];
}

// glob = [mean-pool(o nodes), o context]
__global__ __launch_bounds__(128) void glob_kernel(const float* __restrict__ o,
                                                   float* __restrict__ glob) {
  const int g = blockIdx.x, t = threadIdx.x;
  float s = 0.0f;
  for (int r = 0; r < 32; ++r) s += o[(size_t)(g * 32 + r) * kEmb + t];
  glob[(size_t)g * 256 + t] = s * (1.0f / 32.0f);
  glob[(size_t)g * 256 + 128 + t] = o[(size_t)(kNodes + g) * kEmb + t];
}

// ---------------------------------------------------------------------------
// Host orchestration
// ---------------------------------------------------------------------------
struct Lin { const float* w; const float* b; };
struct LayerP { Lin cscale, e, ff1, ff2, gen, k, lin, q, skip, v; };

static inline int ew_blocks(long long n) { return (int)((n + 255) / 256); }

static void gemm(const float* A, const float* W, const float* b, float* C,
                 int M, int K, int N, int act, bool acc, hipStream_t s) {
  dim3 grid(N / 64, M / 128);
  if (acc)           gemm_wmma_f32<0, true><<<grid, 256, 0, s>>>(A, W, b, C, M, K, N);
  else if (act == 1) gemm_wmma_f32<1, false><<<grid, 256, 0, s>>>(A, W, b, C, M, K, N);
  else               gemm_wmma_f32<0, false><<<grid, 256, 0, s>>>(A, W, b, C, M, K, N);
}

extern "C" void kernel_launch(void* const* d_in, const int* in_sizes, int n_in,
                              void* d_out, int out_size, void* d_ws, size_t ws_size,
                              hipStream_t stream) {
  (void)in_sizes; (void)n_in; (void)out_size; (void)ws_size;

  // ---- unflatten params (jax pytree: sorted dict keys, lin = {b, w}) ----
  int ip = 0;
  auto L = [&](Lin& l) { l.b = (const float*)d_in[ip++]; l.w = (const float*)d_in[ip++]; };
  Lin addedge; L(addedge);
  Lin addnode; L(addnode);
  Lin c2h[3];  L(c2h[0]); L(c2h[1]); L(c2h[2]);
  Lin e2h[3];  L(e2h[0]); L(e2h[1]); L(e2h[2]);
  Lin g_out;   L(g_out);
  LayerP lay[3];
  for (int i = 0; i < 3; ++i) {
    L(lay[i].cscale); L(lay[i].e);   L(lay[i].ff1); L(lay[i].ff2); L(lay[i].gen);
    L(lay[i].k);      L(lay[i].lin); L(lay[i].q);   L(lay[i].skip); L(lay[i].v);
  }
  Lin setedge; L(setedge);
  Lin stopp;   L(stopp);
  Lin x2h[3];  L(x2h[0]); L(x2h[1]); L(x2h[2]);
  const float* x         = (const float*)d_in[ip++];
  const float* edge_attr = (const float*)d_in[ip++];
  const float* cond      = (const float*)d_in[ip++];
  const int*   eidx      = (const int*)d_in[ip++];
  const int*   neidx     = (const int*)d_in[ip++];
  const int*   batch     = (const int*)d_in[ip++];

  // ---- workspace bump allocator ----
  char* wsp = (char*)d_ws;
  auto alloc = [&](size_t bytes) -> void* {
    void* p = wsp; wsp += (bytes + 255) & ~(size_t)255; return p;
  };
  float* o    = (float*)alloc((size_t)kNAug * kEmb * 4);
  float* on   = (float*)alloc((size_t)kNAug * kEmb * 4);   // also loop_attr sum
  float* cbuf = (float*)alloc((size_t)kGraphs * kEmb * 4);
  float* ea   = (float*)alloc((size_t)kEAug * kEmb * 4);
  int*   esrc = (int*)alloc((size_t)kEAug * 4);
  int*   edst = (int*)alloc((size_t)kEAug * 4);
  float* cnt  = (float*)alloc((size_t)kNAug * 4);
  float* xc   = (float*)alloc((size_t)kNAug * 256 * 4);
  float* qb   = (float*)alloc((size_t)kNAug * 256 * 4);
  float* kb   = (float*)alloc((size_t)kNAug * 256 * 4);
  float* vb   = (float*)alloc((size_t)kNAug * 256 * 4);
  float* BIG  = (float*)alloc((size_t)kEAug * 256 * 4);    // ee / t512 / mlp ping-pong
  float* attn = (float*)alloc((size_t)kNAug * 256 * 4);
  float* agg  = (float*)alloc((size_t)kNAug * kEmb * 4);
  float* lh   = (float*)alloc((size_t)kNAug * kEmb * 4);
  float* cs   = (float*)alloc((size_t)kGraphs * 256 * 4);
  float* glob = (float*)alloc((size_t)kGraphs * 256 * 4);
  float* logits = (float*)alloc((size_t)kEAug * 2 * 4);
  unsigned* mx  = (unsigned*)alloc((size_t)kNAug * 2 * 4);
  float* ssum   = (float*)alloc((size_t)kNAug * 2 * 4);
  float* ping = BIG;
  float* pong = BIG + (size_t)kEdges * kEmb;
  float* ee   = BIG;
  float* t512 = BIG;

  float* out_stop    = (float*)d_out;
  float* out_addnode = out_stop + kGraphs;
  float* out_addedge = out_addnode + (size_t)kNodes * 64;
  float* out_setedge = out_addedge + kNonEd;
  float* out_gout    = out_setedge + (size_t)kNonEd * 8;

  // ---- setup: augmented graph ----
  build_edges_kernel<<<ew_blocks(kEdges), 256, 0, stream>>>(eidx, batch, esrc, edst);
  init_ep_kernel<<<ew_blocks((long long)2 * kNodes * kEmb), 256, 0, stream>>>(ea);

  // o = mlp3(x2h, x)
  gemm(x, x2h[0].w, x2h[0].b, ping, kNodes, kXDim, kEmb, 1, false, stream);
  gemm(ping, x2h[1].w, x2h[1].b, pong, kNodes, kEmb, kEmb, 1, false, stream);
  gemm(pong, x2h[2].w, x2h[2].b, o, kNodes, kEmb, kEmb, 0, false, stream);
  // ea[:E] = mlp3(e2h, edge_attr)
  gemm(edge_attr, e2h[0].w, e2h[0].b, ping, kEdges, kEDim, kEmb, 1, false, stream);
  gemm(ping, e2h[1].w, e2h[1].b, pong, kEdges, kEmb, kEmb, 1, false, stream);
  gemm(pong, e2h[2].w, e2h[2].b, ea, kEdges, kEmb, kEmb, 0, false, stream);
  // c = mlp3(c2h, cond); o tail = c
  gemm(cond, c2h[0].w, c2h[0].b, ping, kGraphs, kGDim, kEmb, 1, false, stream);
  gemm(ping, c2h[1].w, c2h[1].b, pong, kGraphs, kEmb, kEmb, 1, false, stream);
  gemm(pong, c2h[2].w, c2h[2].b, cbuf, kGraphs, kEmb, kEmb, 0, false, stream);
  hipMemcpyAsync(o + (size_t)kNodes * kEmb, cbuf, (size_t)kGraphs * kEmb * 4,
                 hipMemcpyDeviceToDevice, stream);

  // loop_attr (mean incoming attr per augmented node) -> ea self-loop rows
  fill_f32_kernel<<<ew_blocks((long long)kNAug * kEmb), 256, 0, stream>>>(on, 0.0f, (long long)kNAug * kEmb);
  fill_f32_kernel<<<ew_blocks(kNAug), 256, 0, stream>>>(cnt, 0.0f, kNAug);
  loopattr_sum_kernel<<<ew_blocks((long long)kEPre * kEmb), 256, 0, stream>>>(ea, edst, on, cnt);
  loopattr_write_kernel<<<ew_blocks((long long)kNAug * kEmb), 256, 0, stream>>>(on, cnt, ea);

  const long long nNF = (long long)kNAug * kEmb;
  const long long nN2 = (long long)kNAug * 256;
  const long long nEF = (long long)kEAug * kEmb;
  const long long nE2 = (long long)kEAug * 256;

  // ---- transformer layers ----
  for (int li = 0; li < 3; ++li) {
    const LayerP& lp = lay[li];
    gemm(cbuf, lp.cscale.w, lp.cscale.b, cs, kGraphs, kEmb, 256, 0, false, stream);
    graph_ln_kernel<<<kGraphs, 128, 0, stream>>>(o, on);
    // message aggregation + gen
    fill_f32_kernel<<<ew_blocks(nNF), 256, 0, stream>>>(agg, 0.0f, nNF);
    msg_agg_kernel<<<ew_blocks(nEF), 256, 0, stream>>>(on, ea, esrc, edst, agg);
    add_inplace_kernel<<<ew_blocks(nNF), 256, 0, stream>>>(agg, on, nNF);
    gemm(agg, lp.gen.w, lp.gen.b, lh, kNAug, kEmb, kEmb, 0, false, stream);
    concat_kernel<<<ew_blocks(nN2), 256, 0, stream>>>(on, lh, xc);
    // projections
    gemm(xc, lp.q.w, lp.q.b, qb, kNAug, 256, 256, 0, false, stream);
    gemm(xc, lp.k.w, lp.k.b, kb, kNAug, 256, 256, 0, false, stream);
    gemm(xc, lp.v.w, lp.v.b, vb, kNAug, 256, 256, 0, false, stream);
    gemm(ea, lp.e.w, lp.e.b, ee, kEAug, kEmb, 256, 0, false, stream);
    // attention
    attn_logit_kernel<<<ew_blocks((long long)kEAug * 2 * 32), 256, 0, stream>>>(qb, kb, ee, esrc, edst, logits);
    fill_u32_kernel<<<ew_blocks(kNAug * 2), 256, 0, stream>>>(mx, 0u, kNAug * 2);
    seg_max_kernel<<<ew_blocks((long long)kEAug * 2), 256, 0, stream>>>(logits, edst, mx);
    fill_f32_kernel<<<ew_blocks(kNAug * 2), 256, 0, stream>>>(ssum, 0.0f, kNAug * 2);
    fill_f32_kernel<<<ew_blocks(nN2), 256, 0, stream>>>(attn, 0.0f, nN2);
    attn_accum_kernel<<<ew_blocks(nE2), 256, 0, stream>>>(logits, mx, vb, ee, esrc, edst, ssum, attn);
    attn_norm_kernel<<<ew_blocks(nN2), 256, 0, stream>>>(attn, ssum);
    // skip + out projection + conditional scale/shift
    gemm(xc, lp.skip.w, lp.skip.b, attn, kNAug, 256, 256, 0, true, stream);
    gemm(attn, lp.lin.w, lp.lin.b, lh, kNAug, 256, kEmb, 0, false, stream);
    o_update_kernel<<<ew_blocks(nNF), 256, 0, stream>>>(o, lh, cs);
    // feed-forward
    graph_ln_kernel<<<kGraphs, 128, 0, stream>>>(o, on);
    gemm(on, lp.ff1.w, lp.ff1.b, t512, kNAug, kEmb, 512, 1, false, stream);
    gemm(t512, lp.ff2.w, lp.ff2.b, o, kNAug, 512, kEmb, 0, true, stream);
  }

  // ---- heads ----
  glob_kernel<<<kGraphs, 128, 0, stream>>>(o, glob);
  rowdot_kernel<1><<<ew_blocks((long long)kGraphs * 32), 256, 0, stream>>>(
      glob, stopp.w, stopp.b, out_stop, kGraphs, 256);
  gemm(o, addnode.w, addnode.b, out_addnode, kNodes, kEmb, 64, 0, false, stream);
  pairdot_kernel<1><<<ew_blocks((long long)kNonEd * 32), 256, 0, stream>>>(
      o, neidx, 1, neidx + kNonEd, 1, addedge.w, addedge.b, out_addedge, kNonEd);
  pairdot_kernel<8><<<ew_blocks((long long)kNonEd * 32), 256, 0, stream>>>(
      o, eidx, 2, eidx + kEdges, 2, setedge.w, setedge.b, out_setedge, kNonEd);
  rowdot_kernel<1><<<ew_blocks((long long)kGraphs * 32), 256, 0, stream>>>(
      glob, g_out.w, g_out.b, out_gout, kGraphs, 256);
}